// EdgeDecoder_67010079752609
// MI455X (gfx1250) — compile-verified
//
#include <hip/hip_runtime.h>

#define NUM_NODES 100000
#define NUM_EDGES 50000
#define NUM_CONN  800000
#define IN_DIM    256
#define HIDDEN    128
#define OUT_DIM   17
#define LN_EPS    1e-5f

#define WAVES_PER_BLOCK 5
#define TILE_ROWS 16
#define N_TILES (HIDDEN / 16)     // 8
#define K_STEPS (IN_DIM / 4)      // 64
// Packed B stream: per K-step, per lane, 8 N-tiles x (K,K+1) pair = 16 floats.
#define WPACK_FLOATS (K_STEPS * 32 * 16)   // 32768 floats = 128 KB

typedef float v2f __attribute__((ext_vector_type(2)));
typedef float v8f __attribute__((ext_vector_type(8)));

// ---------------- kernel 1: zero accumulators ----------------
__global__ void zero_ws_kernel(float* __restrict__ p, int n) {
    int i = blockIdx.x * blockDim.x + threadIdx.x;
    if (i < n) p[i] = 0.0f;
}

// ---------------- kernel 1b: repack W1 into per-lane WMMA B-layout ----------------
// wpack[((ks*32 + lane)*8 + nt)*2 + {0,1}] = W1[(4*ks + 2*hi) + {0,1}][nt*16 + m]
// so the hot loop reads each lane's B operands as 4 contiguous float4 (b128).
__global__ void repack_w1_kernel(const float* __restrict__ W1,
                                 float* __restrict__ wpack) {
    int id = blockIdx.x * blockDim.x + threadIdx.x;   // 0 .. K_STEPS*32*8-1
    if (id >= K_STEPS * 32 * 8) return;
    int ks   = id >> 8;           // /256
    int rem  = id & 255;
    int lane = rem >> 3;
    int nt   = rem & 7;
    int m    = lane & 15;
    int hi   = lane >> 4;
    int krow = ks * 4 + 2 * hi;
    int n    = nt * 16 + m;
    wpack[2 * id + 0] = W1[(size_t)(krow + 0) * HIDDEN + n];
    wpack[2 * id + 1] = W1[(size_t)(krow + 1) * HIDDEN + n];
}

// ---------------- kernel 2: gather + segment scatter-add ----------------
// 32 lanes per connection; each lane moves 8 floats via two float4 loads and
// 8 f32 global atomics (L2-resident accumulator). counts[] bumped by lane 0.
__global__ __launch_bounds__(256) void scatter_add_kernel(
    const float* __restrict__ nf,
    const int*   __restrict__ nodes,
    const int*   __restrict__ edges,
    float* __restrict__ summed,
    float* __restrict__ counts) {
    int t    = blockIdx.x * blockDim.x + threadIdx.x;
    int conn = t >> 5;
    if (conn >= NUM_CONN) return;
    int lane = t & 31;
    int node = nodes[conn];
    int edge = edges[conn];

    const float4* src = (const float4*)(nf + (size_t)node * IN_DIM + lane * 8);
    float4 a = src[0];
    float4 b = src[1];

    float* dst = summed + (size_t)edge * IN_DIM + lane * 8;
    atomicAdd(dst + 0, a.x);
    atomicAdd(dst + 1, a.y);
    atomicAdd(dst + 2, a.z);
    atomicAdd(dst + 3, a.w);
    atomicAdd(dst + 4, b.x);
    atomicAdd(dst + 5, b.y);
    atomicAdd(dst + 6, b.z);
    atomicAdd(dst + 7, b.w);
    if (lane == 0) atomicAdd(counts + edge, 1.0f);
}

// ---------------- kernel 3: fused mean + MLP + LN + ReLU + GEMM2 ----------------
// One wave per 16-edge tile. GEMM1 (256->128) on V_WMMA_F32_16X16X4_F32 with
// full-fp32 accumulation; B operands stream from the packed layout as b128s.
__global__ __launch_bounds__(WAVES_PER_BLOCK * 32) void edge_mlp_kernel(
    const float* __restrict__ summed,
    const float* __restrict__ counts,
    const float* __restrict__ wpack,  // packed W1 (B-layout stream)
    const float* __restrict__ b1,     // [HIDDEN]
    const float* __restrict__ lng,    // [HIDDEN]
    const float* __restrict__ lnb,    // [HIDDEN]
    const float* __restrict__ W2,     // [HIDDEN][OUT_DIM] row-major
    const float* __restrict__ b2,     // [OUT_DIM]
    float* __restrict__ out) {        // [NUM_EDGES][OUT_DIM]
    __shared__ float sH[WAVES_PER_BLOCK * TILE_ROWS * HIDDEN];

    const int wave = threadIdx.x >> 5;
    const int lane = threadIdx.x & 31;
    const int tile = blockIdx.x * WAVES_PER_BLOCK + wave;
    const int m0   = tile * TILE_ROWS;          // 50000 % 16 == 0: always in range
    const int m    = lane & 15;
    const int hi   = lane >> 4;

    const int row = m0 + m;
    const float invc = 1.0f / fmaxf(counts[row], 1.0f);   // segment-mean scale
    const float* __restrict__ arow = summed + (size_t)row * IN_DIM;

    v8f acc[N_TILES] = {};   // 8 x (16x16 f32) accumulators = H[16][128] tile

    // GEMM1: H = (summed * invc) @ W1, K = 256 in steps of 4.
    for (int ks = 0; ks < K_STEPS; ++ks) {
        const int k = ks * 4;
        v2f a = *(const v2f*)(arow + k + 2 * hi);   // contiguous b64 per lane
        a.x *= invc;
        a.y *= invc;

        // 16 contiguous floats of packed B per lane -> 4 x global_load_b128.
        const float4* __restrict__ bp =
            (const float4*)(wpack + ((size_t)ks * 32 + lane) * 16);
        float4 q0 = bp[0], q1 = bp[1], q2 = bp[2], q3 = bp[3];
        v2f bbs[N_TILES];
        bbs[0].x = q0.x; bbs[0].y = q0.y;
        bbs[1].x = q0.z; bbs[1].y = q0.w;
        bbs[2].x = q1.x; bbs[2].y = q1.y;
        bbs[3].x = q1.z; bbs[3].y = q1.w;
        bbs[4].x = q2.x; bbs[4].y = q2.y;
        bbs[5].x = q2.z; bbs[5].y = q2.w;
        bbs[6].x = q3.x; bbs[6].y = q3.y;
        bbs[7].x = q3.z; bbs[7].y = q3.w;

#pragma unroll
        for (int nt = 0; nt < N_TILES; ++nt) {
            acc[nt] = __builtin_amdgcn_wmma_f32_16x16x4_f32(
                false, a, false, bbs[nt], (short)0, acc[nt], false, false);
        }
    }

    // Bias: column of acc[nt] is n = nt*16 + m (same for all 8 rows in the v8f).
#pragma unroll
    for (int nt = 0; nt < N_TILES; ++nt) {
        const float bias = b1[nt * 16 + m];
#pragma unroll
        for (int r = 0; r < 8; ++r) acc[nt][r] += bias;
    }

    // LayerNorm row statistics. D-layout: acc[nt][r] is row (r + 8*hi),
    // column (m + 16*nt); each row lives across the 16 lanes of a half-wave.
    float s[8], q[8];
#pragma unroll
    for (int r = 0; r < 8; ++r) { s[r] = 0.0f; q[r] = 0.0f; }
#pragma unroll
    for (int nt = 0; nt < N_TILES; ++nt) {
#pragma unroll
        for (int r = 0; r < 8; ++r) {
            const float x = acc[nt][r];
            s[r] += x;
            q[r] += x * x;
        }
    }
#pragma unroll
    for (int mask = 1; mask < 16; mask <<= 1) {
#pragma unroll
        for (int r = 0; r < 8; ++r) {
            s[r] += __shfl_xor(s[r], mask, 32);
            q[r] += __shfl_xor(q[r], mask, 32);
        }
    }
    float mean[8], rstd[8];
#pragma unroll
    for (int r = 0; r < 8; ++r) {
        mean[r] = s[r] * (1.0f / HIDDEN);
        const float var = q[r] * (1.0f / HIDDEN) - mean[r] * mean[r];
        rstd[r] = rsqrtf(var + LN_EPS);
    }

    // Normalize + gamma/beta + ReLU, spill tile to LDS for the 128->17 GEMM.
    float* __restrict__ hbase = sH + wave * (TILE_ROWS * HIDDEN);
#pragma unroll
    for (int nt = 0; nt < N_TILES; ++nt) {
        const int n = nt * 16 + m;
        const float g  = lng[n];
        const float bb = lnb[n];
#pragma unroll
        for (int r = 0; r < 8; ++r) {
            float x = (acc[nt][r] - mean[r]) * rstd[r] * g + bb;
            x = fmaxf(x, 0.0f);
            hbase[(r + 8 * hi) * HIDDEN + n] = x;
        }
    }
    __syncthreads();

    // GEMM2: out[16][17] = H[16][128] @ W2[128][17] + b2. 272 outputs / 32 lanes.
    for (int idx = lane; idx < TILE_ROWS * OUT_DIM; idx += 32) {
        const int r = idx / OUT_DIM;
        const int j = idx - r * OUT_DIM;
        float sum = b2[j];
        const float* __restrict__ hr = hbase + r * HIDDEN;
#pragma unroll 4
        for (int k = 0; k < HIDDEN; ++k)
            sum = fmaf(hr[k], W2[k * OUT_DIM + j], sum);
        out[(size_t)(m0 + r) * OUT_DIM + j] = sum;
    }
}

// ---------------- launch ----------------
extern "C" void kernel_launch(void* const* d_in, const int* in_sizes, int n_in,
                              void* d_out, int out_size, void* d_ws, size_t ws_size,
                              hipStream_t stream) {
    const float* nf    = (const float*)d_in[0];
    const int*   hidx  = (const int*)d_in[1];     // [2][NUM_CONN] int32
    const int*   nodes = hidx;
    const int*   edges = hidx + NUM_CONN;
    const float* W1    = (const float*)d_in[2];
    const float* b1    = (const float*)d_in[3];
    const float* lng   = (const float*)d_in[4];
    const float* lnb   = (const float*)d_in[5];
    const float* W2    = (const float*)d_in[6];
    const float* b2    = (const float*)d_in[7];
    float*       out   = (float*)d_out;

    float* summed = (float*)d_ws;                           // [NUM_EDGES][IN_DIM]
    float* counts = summed + (size_t)NUM_EDGES * IN_DIM;    // [NUM_EDGES]
    float* wpack  = counts + NUM_EDGES;                     // [WPACK_FLOATS]

    // Repack W1 into the WMMA B-layout stream (independent of scatter phase).
    const int npack = K_STEPS * 32 * 8;                     // 16384 threads
    repack_w1_kernel<<<(npack + 255) / 256, 256, 0, stream>>>(W1, wpack);

    const int nzero = NUM_EDGES * IN_DIM + NUM_EDGES;
    zero_ws_kernel<<<(nzero + 255) / 256, 256, 0, stream>>>(summed, nzero);

    const long long nthreads = (long long)NUM_CONN * 32;
    scatter_add_kernel<<<(int)((nthreads + 255) / 256), 256, 0, stream>>>(
        nf, nodes, edges, summed, counts);

    const int ntiles = NUM_EDGES / TILE_ROWS;               // 3125 = 5^5
    edge_mlp_kernel<<<ntiles / WAVES_PER_BLOCK, WAVES_PER_BLOCK * 32, 0, stream>>>(
        summed, counts, wpack, b1, lng, lnb, W2, b2, out);
}